// Snake_44427141709947
// MI455X (gfx1250) — compile-verified
//
#include <hip/hip_runtime.h>
#include <stdint.h>

#ifndef __has_builtin
#define __has_builtin(x) 0
#endif
#if __has_builtin(__builtin_amdgcn_global_load_async_to_lds_b128)
#define HAVE_ASYNC_B128 1
#else
#define HAVE_ASYNC_B128 0
#endif
#if __has_builtin(__builtin_amdgcn_s_wait_asynccnt)
#define HAVE_WAIT_ASYNC 1
#endif

typedef __attribute__((ext_vector_type(2))) float v2f;
typedef __attribute__((ext_vector_type(4))) float v4f;
typedef __attribute__((ext_vector_type(8))) float v8f;
typedef __attribute__((ext_vector_type(4))) int   v4i;
typedef v4i __attribute__((address_space(1)))* gv4i;
typedef v4i __attribute__((address_space(3)))* lv4i;

// ---------------- problem constants ----------------
#define BATCH   16
#define CCH     256
#define L_IN    8192
#define L2X     16384
#define NROWS   4096

// ---------------- tiling ----------------
#define WG        256
#define CHG       16               // channels per workgroup (WMMA N dim)
#define TILE_OUT  512              // outputs per channel per tile
#define SEGS      8                // row split across WGs
#define NT        2                // tiles per WG: SEGS*NT*TILE_OUT = 8192
#define NGRP      (NROWS / CHG)    // 256
#define GRID      (NGRP * SEGS)    // 2048

#define XPITCH    548              // floats per channel row of x tile (16B rows)
#define XV4R      137              // float4 per x row
#define XV4T      (CHG * XV4R)     // 2192 float4 per tile
#define XTAIL     (XV4T - 8 * WG)  // 144
#define SPITCH    1060             // floats per channel row of s tile
#define NBLK1     33               // stage-1 16-position i-blocks per tile
#define NBLK2     32               // stage-2 16-position out blocks per tile

// ---------------- CDNA5 helpers ----------------
__device__ __forceinline__ void async_b128(const float* g, float* l) {
#if HAVE_ASYNC_B128
  __builtin_amdgcn_global_load_async_to_lds_b128(
      (gv4i)(uintptr_t)g, (lv4i)(uint32_t)(uintptr_t)l, 0, 0);
#else
  unsigned lds_off = (unsigned)(uintptr_t)l;
  asm volatile("global_load_async_to_lds_b128 %0, %1, off"
               :: "v"(lds_off), "v"(g) : "memory");
#endif
}

template <int N>
__device__ __forceinline__ void wait_asynccnt() {
#if HAVE_WAIT_ASYNC
  __builtin_amdgcn_s_wait_asynccnt(N);
#else
  asm volatile("s_wait_asynccnt %0" :: "n"(N) : "memory");
#endif
}

__device__ __forceinline__ v8f wmma4(v2f a, v2f b, v8f c) {
  return __builtin_amdgcn_wmma_f32_16x16x4_f32(false, a, false, b, (short)0, c,
                                               false, false);
}

// 9 uniform async b128 loads per thread (per-wave ASYNCcnt += 9)
__device__ __forceinline__ void prefetch16(const float* __restrict__ x,
                                           int rowBase, int xbg, float* dst,
                                           float* dump, int tid) {
#pragma unroll
  for (int it = 0; it < 9; ++it) {
    int v  = (it < 8) ? (tid + WG * it) : (8 * WG + (tid % XTAIL));
    int r  = v / XV4R;
    int c4 = v - r * XV4R;
    int gx = xbg + 4 * c4;
    bool ok = (gx >= 0) && (gx <= L_IN - 4);
    const float* gp = x + (size_t)(rowBase + r) * L_IN + (ok ? gx : 0);
    float*       lp = ok ? (dst + r * XPITCH + 4 * c4) : dump;
    async_b128(gp, lp);
  }
}

// ---------------- fused WMMA snake kernel ----------------
__global__ __launch_bounds__(WG) void Snake_44427141709947_kernel(
    const float* __restrict__ x, const float* __restrict__ alpha,
    const float* __restrict__ filt, float* __restrict__ out) {
  __shared__ __align__(16) float xs[2][CHG * XPITCH];  // x tiles / out staging
  __shared__ __align__(16) float ss[CHG * SPITCH];     // post-snake z tile
  __shared__ __align__(16) float lF[16];
  __shared__ __align__(16) float dump[4];

  const int tid  = threadIdx.x;
  const int lane = tid & 31;
  const int wv   = tid >> 5;        // wave 0..7 (wave32)
  const int half = lane >> 4;       // VGPR half (K split for A/B layouts)
  const int nI   = lane & 15;       // A row (M) / B,D column (N = channel)
  const int grp  = blockIdx.x & (NGRP - 1);
  const int seg  = blockIdx.x >> 8;
  const int rowBase = grp * CHG;
  const int segBase = seg * (NT * TILE_OUT);

  // zero x buffers once (zero-coeff WMMA columns must never see NaN bits)
  for (int i = tid; i < 2 * CHG * XPITCH; i += WG) ((float*)xs)[i] = 0.0f;
  if (tid < 16) lF[tid] = (tid < 12) ? filt[tid] : 0.0f;
  __syncthreads();

  // ---- per-lane constant banded A matrices (16x4 f32: VGPR0=K0|K2, VGPR1=K1|K3)
  const int kA0 = 2 * half, kA1 = kA0 + 1;
  v2f aE[6], aO[6], aD[11];
#pragma unroll
  for (int c = 0; c < 6; ++c) {     // stage-1: z_e/z_o[m] = sum Fe/Fo[m'] x[...]
    int d0 = 4 * c + kA0 - nI, d1 = 4 * c + kA1 - nI;
    v2f e, o;
    e.x = (d0 >= 0 && d0 < 6) ? 2.0f * lF[2 * d0 + 1] : 0.0f;
    e.y = (d1 >= 0 && d1 < 6) ? 2.0f * lF[2 * d1 + 1] : 0.0f;
    o.x = (d0 >= 0 && d0 < 6) ? 2.0f * lF[2 * d0] : 0.0f;
    o.y = (d1 >= 0 && d1 < 6) ? 2.0f * lF[2 * d1] : 0.0f;
    aE[c] = e; aO[c] = o;
  }
#pragma unroll
  for (int c = 0; c < 11; ++c) {    // stage-2: out[m] = sum F[k] s[2m+k]
    int e0 = 4 * c + kA0 - 2 * nI, e1 = 4 * c + kA1 - 2 * nI;
    v2f d;
    d.x = (e0 >= 0 && e0 < 12) ? lF[e0] : 0.0f;
    d.y = (e1 >= 0 && e1 < 12) ? lF[e1] : 0.0f;
    aD[c] = d;
  }
  const float a    = alpha[((grp & 15) << 4) + nI];
  const float ainv = 1.0f / (a + 1e-9f);

  prefetch16(x, rowBase, segBase - 16, xs[0], dump, tid);

  for (int t = 0; t < NT; ++t) {
    const int o0  = segBase + t * TILE_OUT;
    const int cur = t & 1;
    const int sb  = 2 * o0 - 8;       // global j of ss[..][0]
    float* xb  = xs[cur];
    float* ob_ = xs[cur];             // reused as output staging in stage 2/3

    if (t) __syncthreads();           // prev tile's staging reads complete
    if (t + 1 < NT) {
      prefetch16(x, rowBase, segBase + (t + 1) * TILE_OUT - 16, xs[cur ^ 1],
                 dump, tid);
      wait_asynccnt<9>();             // tile t's 9 loads done (in-order)
    } else {
      wait_asynccnt<0>();
    }
    __syncthreads();                  // x tile visible to all waves

    // ---- stage 1: polyphase upsample (WMMA) + snake -> ss ----
    for (int blk = wv; blk < NBLK1; blk += 8) {
      const int qb = 16 * blk + 10 + 2 * half;   // x float offset of B K-row
      const float* bp = xb + nI * XPITCH + qb;
      v2f b[6];
#pragma unroll
      for (int c = 0; c < 6; ++c) b[c] = *(const v2f*)(bp + 4 * c);
      v8f de = {}, dd = {};
#pragma unroll
      for (int c = 0; c < 6; ++c) {
        de = wmma4(aE[c], b[c], de);
        dd = wmma4(aO[c], b[c], dd);
      }
      // D[m,n]: lane half picks M = v + 8*half, N = nI (channel)
      float* sp = ss + nI * SPITCH + (32 * blk + 16 * half);
#pragma unroll
      for (int v = 0; v < 8; ++v) {
        float ze = de[v], zo = dd[v];
        float se = __sinf(a * ze), so = __sinf(a * zo);
        v2f pr;
        pr.x = ze + ainv * (se * se);
        pr.y = zo + ainv * (so * so);
        *(v2f*)(sp + 2 * v) = pr;     // interleaved even/odd j
      }
    }
    // exact row-edge z (replicate 2*x[0] left pad, zero right pad); regions
    // overwritten belong to wave 0's own blocks (0 and 32) -> in-order LDS
    if (wv == 0 && (o0 == 0 || o0 == L_IN - TILE_OUT)) {
      const int jb = (o0 == 0) ? 0 : (L2X - 8);
#pragma unroll
      for (int cc = 0; cc < 4; ++cc) {
        int ch = (lane >> 3) + 4 * cc;
        int j  = jb + (lane & 7);
        const float* xr = xb + ch * XPITCH;
        float x0v = xr[16];           // global x[o0-16+16]; == x[0] when o0==0
        float acc = 0.0f;
#pragma unroll
        for (int k = 0; k < 12; ++k) {
          int u = j + k - 5;
          float vv;
          if (u < 0)         vv = 2.0f * x0v;
          else if (u >= L2X) vv = 0.0f;
          else if (u & 1)    vv = 0.0f;
          else               vv = 2.0f * xr[(u >> 1) - (o0 - 16)];
          acc += lF[k] * vv;
        }
        float aa = alpha[((grp & 15) << 4) + ch];
        float sv = __sinf(aa * acc);
        ss[ch * SPITCH + (j - sb)] = acc + (1.0f / (aa + 1e-9f)) * sv * sv;
      }
    }
    __syncthreads();

    // ---- stage 2: stride-2 downsample (WMMA) -> ob_ staging ----
    for (int ob = wv; ob < NBLK2; ob += 8) {
      const int qb2 = 32 * ob + 3 + 2 * half;    // ss float offset of B K-row
      const float* bp2 = ss + nI * SPITCH + qb2;
      v8f d2 = {};
#pragma unroll
      for (int c = 0; c < 11; ++c) {
        v2f b2;                       // odd offset -> two b32 loads
        b2.x = bp2[4 * c];
        b2.y = bp2[4 * c + 1];
        d2 = wmma4(aD[c], b2, d2);
      }
      float* op = ob_ + nI * XPITCH + 16 * ob + 8 * half;
#pragma unroll
      for (int v = 0; v < 8; ++v) op[v] = d2[v];
    }
    // exact edge outputs (downsample clamp pad): o in {0,1,2} / {L-3..L-1};
    // regions belong to wave 0 (block 0) / wave 7 (block 31) respectively
    if (wv == 0 && o0 == 0) {
#pragma unroll
      for (int cc = 0; cc < 2; ++cc) {
        int idx = lane + 32 * cc, ch = idx >> 2, oo = idx & 3;
        if (oo < 3) {
          float acc = 0.0f;
#pragma unroll
          for (int k = 0; k < 12; ++k) {
            int j = 2 * oo - 5 + k;
            j = j < 0 ? 0 : j;
            acc += lF[k] * ss[ch * SPITCH + (j - sb)];
          }
          ob_[ch * XPITCH + oo] = acc;
        }
      }
    }
    if (wv == 7 && o0 == L_IN - TILE_OUT) {
#pragma unroll
      for (int cc = 0; cc < 2; ++cc) {
        int idx = lane + 32 * cc, ch = idx >> 2, oo = idx & 3;
        if (oo < 3) {
          int o = L_IN - 3 + oo;
          float acc = 0.0f;
#pragma unroll
          for (int k = 0; k < 12; ++k) {
            int j = 2 * o - 5 + k;
            j = j > L2X - 1 ? L2X - 1 : j;
            acc += lF[k] * ss[ch * SPITCH + (j - sb)];
          }
          ob_[ch * XPITCH + (o - o0)] = acc;
        }
      }
    }
    __syncthreads();

    // ---- stage 3: coalesced b128 global stores ----
#pragma unroll
    for (int w = 0; w < 8; ++w) {
      int i4 = tid + WG * w;          // 16 ch * 128 float4
      int r  = i4 >> 7;
      int p4 = (i4 & 127) << 2;
      v4f val = *(const v4f*)(ob_ + r * XPITCH + p4);
      *(v4f*)(out + (size_t)(rowBase + r) * L_IN + o0 + p4) = val;
    }
  }
}

// ---------------- launcher ----------------
extern "C" void kernel_launch(void* const* d_in, const int* in_sizes, int n_in,
                              void* d_out, int out_size, void* d_ws,
                              size_t ws_size, hipStream_t stream) {
  const float* x     = (const float*)d_in[0];
  const float* alpha = (const float*)d_in[1];
  const float* filt  = (const float*)d_in[2];
  float*       out   = (float*)d_out;
  (void)in_sizes; (void)n_in; (void)out_size; (void)d_ws; (void)ws_size;

  dim3 grid(GRID), block(WG);
  hipLaunchKernelGGL(Snake_44427141709947_kernel, grid, block, 0, stream, x,
                     alpha, filt, out);
}